// GCN_39298950758977
// MI455X (gfx1250) — compile-verified
//
#include <hip/hip_runtime.h>
#include <hip/hip_bf16.h>

#define NPG   128     // nodes per graph
#define EPG   2048    // edges per graph
#define IN_CH 151
#define HID   64
#define THREADS 256

typedef float v2f __attribute__((ext_vector_type(2)));
typedef float v8f __attribute__((ext_vector_type(8)));

__device__ __forceinline__ v8f wmma_f32_16x16x4(v2f a, v2f b, v8f c) {
    // 8 args: (neg_a, A, neg_b, B, c_mod, C, reuse_a, reuse_b)
    return __builtin_amdgcn_wmma_f32_16x16x4_f32(false, a, false, b, (short)0, c, false, false);
}

// One workgroup per graph. Fully fused GCNConv + readout:
//   h = x_g @ W1 (WMMA, f32)          -> LDS
//   deg/dinv via LDS atomics          -> LDS
//   out[col] += dinv[row]*w*dinv[col]*h[row]  (wave-per-edge ds_add_f32)
//   res = sigmoid( sum relu(out + b1) * Wlin + blin )
__global__ __launch_bounds__(THREADS)
void gcn_fused_kernel(const float* __restrict__ x,
                      const int*   __restrict__ rowIdx,
                      const int*   __restrict__ colIdx,
                      const float* __restrict__ ew,
                      const float* __restrict__ W1,
                      const float* __restrict__ b1,
                      const float* __restrict__ Wlin,
                      const float* __restrict__ blin,
                      float*       __restrict__ out)
{
    // ---- LDS layout (single pool, region reuse) -------------------------
    // [0      , 9728 )  ws   : W1 padded to 152 rows x 64     (38.9 KB)
    // [0      , 8192 )  outA : aggregation buffer, REUSES ws after GEMM
    // [9728   , 17920)  h    : 128 x 64 f32                   (32 KB)
    // [17920  , 18048)  deg  : degree -> dinv (128)
    // [18048  , 18112)  b1s  : bias (64)
    // [18112  , 18120)  red  : per-wave reduction (8)
    __shared__ float smem[18120];
    float* ws   = smem;
    float* outA = smem;            // alias: valid after GEMM barrier
    float* hL   = smem + 9728;
    float* deg  = smem + 17920;
    float* b1s  = smem + 18048;
    float* red  = smem + 18112;

    const int tid  = threadIdx.x;
    const int lane = tid & 31;
    const int wid  = tid >> 5;     // 0..7 (8 wave32s)
    const int l16  = lane & 15;
    const int hi   = lane >> 4;    // 0 or 1 (half-wave)
    const int g    = blockIdx.x;
    const int nodeBase = g * NPG;
    const long eBase   = (long)g * EPG;

    // ---- Phase 1: stage W1 (+zero pad row 151), bias, init degrees ------
    for (int i = tid; i < IN_CH * HID; i += THREADS) ws[i] = W1[i];
    if (tid < HID) { ws[IN_CH * HID + tid] = 0.0f; b1s[tid] = b1[tid]; }
    if (tid < NPG) deg[tid] = 1.0f;                 // self-loop weight
    __syncthreads();

    // ---- Phase 2: degree accumulation (LDS atomics) ---------------------
    for (int e = tid; e < EPG; e += THREADS) {
        int c = colIdx[eBase + e] - nodeBase;
        atomicAdd(&deg[c], ew[eBase + e]);
    }
    __syncthreads();
    if (tid < NPG) deg[tid] = rsqrtf(deg[tid]);     // deg >= 1 always

    // ---- Phase 3: h = x_g @ W1 via V_WMMA_F32_16X16X4_F32 ---------------
    // Wave w owns rows [16w,16w+16); A fragments streamed from global
    // (each lane sweeps its row linearly -> cacheline reuse in L0),
    // B fragments from LDS-staged W1.
    const float* pA = x + (long)(nodeBase + wid * 16 + l16) * IN_CH;
    const int koff = hi * 2;       // half-wave K offset per ISA layout
    const int n0 = l16, n1 = 16 + l16, n2 = 32 + l16, n3 = 48 + l16;

    v8f a0 = {}, a1 = {}, a2 = {}, a3 = {};
    for (int k0 = 0; k0 < 148; k0 += 4) {           // k = 0..147
        const int k = k0 + koff;
        v2f af; af.x = pA[k]; af.y = pA[k + 1];
        const float* wr = ws + k * HID;             // rows k, k+1
        v2f bf;
        bf.x = wr[n0]; bf.y = wr[HID + n0]; a0 = wmma_f32_16x16x4(af, bf, a0);
        bf.x = wr[n1]; bf.y = wr[HID + n1]; a1 = wmma_f32_16x16x4(af, bf, a1);
        bf.x = wr[n2]; bf.y = wr[HID + n2]; a2 = wmma_f32_16x16x4(af, bf, a2);
        bf.x = wr[n3]; bf.y = wr[HID + n3]; a3 = wmma_f32_16x16x4(af, bf, a3);
    }
    {   // remainder k0 = 148: k in {148,150}; element 151 is zero-padded
        const int k = 148 + koff;
        v2f af; af.x = pA[k]; af.y = (k + 1 < IN_CH) ? pA[k + 1] : 0.0f;
        const float* wr = ws + k * HID;             // ws row 151 == 0
        v2f bf;
        bf.x = wr[n0]; bf.y = wr[HID + n0]; a0 = wmma_f32_16x16x4(af, bf, a0);
        bf.x = wr[n1]; bf.y = wr[HID + n1]; a1 = wmma_f32_16x16x4(af, bf, a1);
        bf.x = wr[n2]; bf.y = wr[HID + n2]; a2 = wmma_f32_16x16x4(af, bf, a2);
        bf.x = wr[n3]; bf.y = wr[HID + n3]; a3 = wmma_f32_16x16x4(af, bf, a3);
    }

    // D layout: VGPR r -> M = r (lanes 0-15) / r+8 (lanes 16-31), N = lane%16
    {
        const int mrow = wid * 16 + hi * 8;
#pragma unroll
        for (int r = 0; r < 8; ++r) {
            hL[(mrow + r) * HID + n0] = a0[r];
            hL[(mrow + r) * HID + n1] = a1[r];
            hL[(mrow + r) * HID + n2] = a2[r];
            hL[(mrow + r) * HID + n3] = a3[r];
        }
    }
    __syncthreads();   // h + dinv ready; ws region now dead -> reuse as outA

    // ---- Phase 4: init out with self-loop term dinv[i]^2 * h[i] ---------
    for (int idx = tid; idx < NPG * HID; idx += THREADS) {
        const float dv = deg[idx >> 6];
        outA[idx] = dv * dv * hL[idx];
    }
    __syncthreads();

    // ---- Phase 5: edge scatter, wave-per-edge (conflict-free ds_add) ----
    {
        const int j2 = lane * 2;
        for (int e = wid; e < EPG; e += 8) {
            const long eg = eBase + e;
            const int  r  = rowIdx[eg] - nodeBase;
            const int  c  = colIdx[eg] - nodeBase;
            const float cf = deg[r] * ew[eg] * deg[c];
            const float h0 = hL[r * HID + j2];
            const float h1 = hL[r * HID + j2 + 1];
            atomicAdd(&outA[c * HID + j2],     cf * h0);
            atomicAdd(&outA[c * HID + j2 + 1], cf * h1);
        }
    }
    __syncthreads();

    // ---- Phase 6: readout GEMV + block reduce + sigmoid -----------------
    float acc = 0.0f;
    for (int idx = tid; idx < NPG * HID; idx += THREADS) {
        float v = outA[idx] + b1s[idx & 63];
        v = v > 0.0f ? v : 0.0f;                    // ReLU
        acc += v * Wlin[idx];
    }
#pragma unroll
    for (int off = 16; off > 0; off >>= 1)
        acc += __shfl_xor(acc, off, 32);            // wave32 reduce
    if (lane == 0) red[wid] = acc;
    __syncthreads();
    if (tid == 0) {
        float s = blin[0];
#pragma unroll
        for (int w8 = 0; w8 < 8; ++w8) s += red[w8];
        out[g] = 1.0f / (1.0f + __expf(-s));
    }
}

extern "C" void kernel_launch(void* const* d_in, const int* in_sizes, int n_in,
                              void* d_out, int out_size, void* d_ws, size_t ws_size,
                              hipStream_t stream) {
    const float* x    = (const float*)d_in[0];
    const int*   ei   = (const int*)d_in[1];        // [2, E]
    const float* ew   = (const float*)d_in[2];
    // d_in[3] = batch (unused: nodes are batch-sorted, 128/graph)
    const float* W1   = (const float*)d_in[4];
    const float* b1   = (const float*)d_in[5];
    const float* Wlin = (const float*)d_in[6];
    const float* blin = (const float*)d_in[7];
    float*       out  = (float*)d_out;

    const int E = in_sizes[1] / 2;                  // 4194304
    const int numGraphs = out_size;                 // 2048
    const int* rowIdx = ei;
    const int* colIdx = ei + E;
    (void)d_ws; (void)ws_size; (void)n_in;

    gcn_fused_kernel<<<numGraphs, THREADS, 0, stream>>>(
        x, rowIdx, colIdx, ew, W1, b1, Wlin, blin, out);
}